// CTCLoss_35605278883990
// MI455X (gfx1250) — compile-verified
//
#include <hip/hip_runtime.h>
#include <stdint.h>

#ifndef NEG_INF
#define NEG_INF (-1e30f)
#endif

// Problem dimensions (fixed by the reference)
#define CTC_B 128
#define CTC_T 2000
#define CTC_V 128
#define CTC_S 200
#define CTC_L (2 * CTC_S + 1)   // 401
#define CTC_AP (2 + CTC_L + 1)  // padded alpha row: 2 front NEG_INF pads + L + tail pad = 404

// ---- gfx1250 async global->LDS copy helpers (ASYNCcnt-tracked DMA path) ----
// VDST VGPR carries the wave-relative LDS byte address; a flat/generic LDS pointer
// keeps that offset in its low 32 bits (FLAT aperture rule), so trunc is correct.
__device__ __forceinline__ void async_copy_b128_to_lds(const float* gptr, const float* lptr) {
    uint32_t lds_addr = (uint32_t)(uintptr_t)lptr;
    uint64_t gaddr    = (uint64_t)(uintptr_t)gptr;
    asm volatile("global_load_async_to_lds_b128 %0, %1, off"
                 :
                 : "v"(lds_addr), "v"(gaddr)
                 : "memory");
}

__device__ __forceinline__ void wait_asynccnt0() {
    asm volatile("s_wait_asynccnt 0" ::: "memory");
}

// 256 threads = 8 waves per workgroup; each 128-thread half owns one batch element
// (2 waves per SIMD32 for latency hiding on the exp/log dependency chain).
// Per half: alpha double-buffered in LDS with a 2-slot NEG_INF front pad (no boundary
// branches on a1/a2); the 512B emission row for step t+1 is DMA'd into the other lp
// buffer by the half's first wave (32 lanes x b128) while the DP update runs.
__global__ __launch_bounds__(256)
void ctc_alpha_kernel(const float* __restrict__ log_probs,  // [B,T,V]
                      const int*   __restrict__ targets,    // [B,S]
                      const int*   __restrict__ input_length,
                      const int*   __restrict__ target_length,
                      float*       __restrict__ per_sample_loss) {
    __shared__ alignas(16) float lp[2][2][CTC_V];   // [half][buf][V]
    __shared__ alignas(16) float A[2][2][CTC_AP];   // [half][buf][padded L]
    __shared__ int tend_sh[2];

    const int tid  = threadIdx.x;
    const int half = tid >> 7;
    const int ltid = tid & 127;
    const int b    = blockIdx.x * 2 + half;

    const float* lpb = log_probs + (size_t)b * CTC_T * CTC_V;

    int my_t_end = input_length[b];
    if (my_t_end > CTC_T) my_t_end = CTC_T;
    const int tl = target_length[b];
    if (ltid == 0) tend_sh[half] = my_t_end;

    // Hoist extended-label indices and skip masks into registers (constant over t).
    int zi[4];
    int allow[4];
#pragma unroll
    for (int k = 0; k < 4; ++k) {
        const int s = ltid + 128 * k;
        int z = 0, al = 0;
        if (s < CTC_L && (s & 1)) {
            const int i = s >> 1;
            z  = targets[b * CTC_S + i];
            al = (s >= 3) ? (z != targets[b * CTC_S + i - 1]) : 1;
        }
        zi[k]    = z;
        allow[k] = al;
    }

    // Async DMA of emission row t=0: first wave of each half, 32 lanes x 16B = 512B.
    if (ltid < 32) {
        async_copy_b128_to_lds(lpb + ltid * 4, &lp[half][0][ltid * 4]);
    }

    // alpha_{-1}: NEG_INF everywhere except state 0 (slot 2 in padded layout).
    for (int s = ltid; s < CTC_AP; s += 128) {
        A[half][0][s] = (s == 2) ? 0.0f : NEG_INF;
        A[half][1][s] = NEG_INF;  // pads of buffer 1 stay NEG_INF forever
    }

    wait_asynccnt0();
    __syncthreads();

    const int t_max = max(tend_sh[0], tend_sh[1]);

    int cur = 0;
    for (int t = 0; t < t_max; ++t) {
        const int nxt = cur ^ 1;

        // DMA next emission row while computing this step (wave-uniform guard).
        if (ltid < 32 && (t + 1) < my_t_end) {
            async_copy_b128_to_lds(lpb + (size_t)(t + 1) * CTC_V + ltid * 4,
                                   &lp[half][nxt][ltid * 4]);
        }

        const bool active = (t < my_t_end);  // reference freeze semantics

#pragma unroll
        for (int k = 0; k < 4; ++k) {
            const int s = ltid + 128 * k;
            if (s < CTC_L) {
                const float a0 = A[half][cur][2 + s];
                const float a1 = A[half][cur][1 + s];
                const float a2 = allow[k] ? A[half][cur][s] : NEG_INF;
                const float m  = fmaxf(a0, fmaxf(a1, a2));
                const float l  = m + __logf(__expf(a0 - m) + __expf(a1 - m) + __expf(a2 - m));
                const float em = lp[half][cur][zi[k]];
                const float na = em + l;
                // Frozen halves must still propagate alpha into the next buffer.
                A[half][nxt][2 + s] = active ? na : a0;
            }
        }

        wait_asynccnt0();   // issuing wave's DMA has landed in LDS
        __syncthreads();    // publish alpha writes + DMA'd row to all waves
        cur = nxt;
    }

    if (ltid == 0) {
        const int   e1 = 2 * tl;       // last blank
        const int   e2 = e1 - 1;       // last label
        const float a1 = A[half][cur][2 + e1];
        const float a2 = A[half][cur][2 + e2];
        const float m  = fmaxf(a1, a2);
        per_sample_loss[b] = -(m + __logf(__expf(a1 - m) + __expf(a2 - m)));
    }
}

// Deterministic mean over the 128 per-sample losses (fixed-order LDS tree reduce).
__global__ __launch_bounds__(128)
void ctc_reduce_kernel(const float* __restrict__ per_sample_loss, float* __restrict__ out) {
    __shared__ float red[128];
    const int tid = threadIdx.x;
    red[tid] = per_sample_loss[tid];
    __syncthreads();
#pragma unroll
    for (int off = 64; off > 0; off >>= 1) {
        if (tid < off) red[tid] += red[tid + off];
        __syncthreads();
    }
    if (tid == 0) out[0] = red[0] / (float)CTC_B;
}

extern "C" void kernel_launch(void* const* d_in, const int* in_sizes, int n_in,
                              void* d_out, int out_size, void* d_ws, size_t ws_size,
                              hipStream_t stream) {
    const float* log_probs     = (const float*)d_in[0];
    const int*   targets       = (const int*)d_in[1];
    const int*   input_length  = (const int*)d_in[2];
    const int*   target_length = (const int*)d_in[3];

    float* per_sample = (float*)d_ws;   // 128 floats of scratch
    float* out        = (float*)d_out;  // single f32 (mean loss)

    ctc_alpha_kernel<<<CTC_B / 2, 256, 0, stream>>>(log_probs, targets, input_length,
                                                    target_length, per_sample);
    ctc_reduce_kernel<<<1, 128, 0, stream>>>(per_sample, out);
}